// GCL_model_41068477284987
// MI455X (gfx1250) — compile-verified
//
#include <hip/hip_runtime.h>

// ---------------- problem constants (from reference) ----------------
#define N_NODES  100000
#define N_EDGES  1600000
#define FEAT     128
#define HID      64
#define N_GRAPHS 256
#define OUT_COLS (3 * HID)   // 192

typedef float v2f __attribute__((ext_vector_type(2)));
typedef float v8f __attribute__((ext_vector_type(8)));

// ---------------- small utility kernels ----------------
__global__ void k_zero_f32(float* __restrict__ p, int n) {
    int i = blockIdx.x * blockDim.x + threadIdx.x;
    if (i < n) p[i] = 0.0f;
}

__global__ void k_deg_init(float* __restrict__ deg) {
    int i = blockIdx.x * blockDim.x + threadIdx.x;
    if (i < N_NODES) deg[i] = 1.0f;              // self-loop contributes 1
}

__global__ void k_deg_count(const int* __restrict__ dst, float* __restrict__ deg) {
    int e = blockIdx.x * blockDim.x + threadIdx.x;
    if (e < N_EDGES) atomicAdd(&deg[dst[e]], 1.0f);
}

__global__ void k_deg_to_dis(float* __restrict__ deg) {
    int i = blockIdx.x * blockDim.x + threadIdx.x;
    if (i < N_NODES) {
        float d = deg[i];
        deg[i] = (d > 0.0f) ? rsqrtf(d) : 0.0f;  // D^{-1/2}
    }
}

// ---------------- fp32 WMMA GEMM with fused self-loop seed ----------------
// C[N,64] = A[N,K] @ W[K,64]; additionally S[i,:] = C[i,:] * dis[i]^2
// (the GCN self-loop message), written straight from the accumulator tile so
// the aggregation seed never has to re-read C from memory.
//
// Block = 128 threads = 4 waves. blockIdx.x selects a 16-row slab of A;
// wave w owns output columns [16w, 16w+16). Each wave accumulates a 16x16
// f32 tile with v_wmma_f32_16x16x4_f32, stepping K by 4.
//
// 32-bit A 16x4 fragment layout (ISA 7.12.2): lanes 0-15 hold K={0,1} for
// row M=lane; lanes 16-31 hold K={2,3} for row M=lane-16. B (4x16) mirrors
// this with N striped across lanes. C/D: VGPR v, lanes 0-15 -> M=v,
// lanes 16-31 -> M=v+8, N=lane%16.
__global__ __launch_bounds__(128)
void k_gemm_wmma_f32(const float* __restrict__ A, const float* __restrict__ W,
                     const float* __restrict__ dis,
                     float* __restrict__ C, float* __restrict__ S, int K) {
    const int wave = threadIdx.x >> 5;
    const int lane = threadIdx.x & 31;
    const int half = lane >> 4;           // 0: K pair {0,1}, 1: K pair {2,3}
    const int l16  = lane & 15;
    const int row0 = blockIdx.x * 16;     // N_NODES % 16 == 0 -> no guard, EXEC all-1s
    const int col  = (wave << 4) + l16;   // output column

    v8f acc = {};
    const float* __restrict__ arow = A + (size_t)(row0 + l16) * K;

    for (int k = 0; k < K; k += 4) {
        const int ka = k + (half << 1);
        v2f a; a.x = arow[ka];
               a.y = arow[ka + 1];
        v2f b; b.x = W[(size_t)ka * HID + col];
               b.y = W[(size_t)(ka + 1) * HID + col];
        acc = __builtin_amdgcn_wmma_f32_16x16x4_f32(
            /*neg_a=*/false, a, /*neg_b=*/false, b,
            /*c_mod=*/(short)0, acc, /*reuse_a=*/false, /*reuse_b=*/false);
    }

    const int rbase = row0 + (half << 3);
    float* __restrict__ cp = C + (size_t)rbase * HID + col;
    float* __restrict__ sp = S + (size_t)rbase * HID + col;
#pragma unroll
    for (int v = 0; v < 8; ++v) {
        float d = dis[rbase + v];                 // broadcast within half-wave
        float val = acc[v];
        cp[(size_t)v * HID] = val;                // linear features (edge gather src)
        sp[(size_t)v * HID] = val * d * d;        // self-loop seed of aggregation
    }
}

// ---------------- edge aggregation: one wave per edge ----------------
// Lane l moves columns {2l, 2l+1}: a single b64 load covers the full 256B row
// coalesced across the wave; two f32 atomics scatter into agg[dst]. Both the
// gathered rows (25.6 MB) and the scatter targets live in the 192 MB L2.
__global__ __launch_bounds__(256)
void k_agg_edges(const int* __restrict__ src, const int* __restrict__ dst,
                 const float* __restrict__ dis, const float* __restrict__ hlin,
                 float* __restrict__ agg) {
    int gid  = blockIdx.x * blockDim.x + threadIdx.x;
    int e    = gid >> 5;
    int lane = gid & 31;
    if (e >= N_EDGES) return;
    int s = src[e], d = dst[e];
    float w = dis[s] * dis[d];
    float2 v = ((const float2*)(hlin + (size_t)s * HID))[lane];
    float* ap = agg + (size_t)d * HID + (lane << 1);
    atomicAdd(ap,     v.x * w);
    atomicAdd(ap + 1, v.y * w);
}

// ---------------- bias + ReLU (in place) + segment pooling ----------------
// batch is sorted, so each thread accumulates NPT consecutive nodes for one
// column and flushes to out[graph*192 + off + c] only when the graph id
// changes -> ~NPT x fewer pooling atomics.
#define NPT 8
__global__ __launch_bounds__(256)
void k_relu_bias_pool(float* __restrict__ h, const float* __restrict__ bias,
                      const int* __restrict__ batch, float* __restrict__ out,
                      int layerOff) {
    const int c   = threadIdx.x & (HID - 1);
    const int grp = threadIdx.x >> 6;                  // 0..3
    const int nodeBase = (blockIdx.x * 4 + grp) * NPT;
    const float b = bias[c];

    float sum = 0.0f;
    int prevG = -1;
    for (int j = 0; j < NPT; ++j) {
        int i = nodeBase + j;
        if (i >= N_NODES) break;
        int g = batch[i];
        if (g != prevG) {
            if (prevG >= 0) atomicAdd(&out[(size_t)prevG * OUT_COLS + layerOff + c], sum);
            sum = 0.0f; prevG = g;
        }
        float v = h[(size_t)i * HID + c] + b;
        v = v > 0.0f ? v : 0.0f;
        h[(size_t)i * HID + c] = v;                    // post-ReLU feature (next layer input)
        sum += v;
    }
    if (prevG >= 0) atomicAdd(&out[(size_t)prevG * OUT_COLS + layerOff + c], sum);
}

// ---------------- host orchestration ----------------
extern "C" void kernel_launch(void* const* d_in, const int* in_sizes, int n_in,
                              void* d_out, int out_size, void* d_ws, size_t ws_size,
                              hipStream_t stream) {
    const float* x     = (const float*)d_in[0];          // [N, 128]
    const int*   ei    = (const int*)  d_in[1];          // [2, E]: row0=src, row1=dst
    const int*   batch = (const int*)  d_in[2];          // [N], sorted
    const float* W[3]  = { (const float*)d_in[3], (const float*)d_in[5], (const float*)d_in[7] };
    const float* B[3]  = { (const float*)d_in[4], (const float*)d_in[6], (const float*)d_in[8] };
    float* out = (float*)d_out;

    const int* srcIdx = ei;
    const int* dstIdx = ei + N_EDGES;

    // workspace: dis [N] | hlin [N,64] | agg/h [N,64]   (~51.6 MB total)
    float* dis  = (float*)d_ws;
    float* hlin = dis  + N_NODES;                        // 400000 B offset, 16B aligned
    float* hbuf = hlin + (size_t)N_NODES * HID;

    // zero the (poisoned) output accumulator
    k_zero_f32<<<(out_size + 255) / 256, 256, 0, stream>>>(out, out_size);

    // symmetric normalization D^{-1/2}
    k_deg_init  <<<(N_NODES + 255) / 256, 256, 0, stream>>>(dis);
    k_deg_count <<<(N_EDGES + 255) / 256, 256, 0, stream>>>(dstIdx, dis);
    k_deg_to_dis<<<(N_NODES + 255) / 256, 256, 0, stream>>>(dis);

    const float* hin = x;
    int K = FEAT;
    for (int L = 0; L < 3; ++L) {
        // hlin = hin @ W[L]; hbuf = hlin * dis^2 (self-loop seed), fused epilogue
        k_gemm_wmma_f32<<<N_NODES / 16, 128, 0, stream>>>(hin, W[L], dis, hlin, hbuf, K);
        // += edge messages (one wave per edge)
        k_agg_edges<<<(N_EDGES * 32) / 256, 256, 0, stream>>>(srcIdx, dstIdx, dis, hlin, hbuf);
        // bias + ReLU in place, pool into out[:, L*64 : L*64+64]
        k_relu_bias_pool<<<(N_NODES + 4 * NPT - 1) / (4 * NPT), 256, 0, stream>>>(
            hbuf, B[L], batch, out, L * HID);
        hin = hbuf;   // post-ReLU features feed the next layer
        K = HID;      // layers 2,3: K = 64
    }
}